// OptimalTransportLayer_74620761801307
// MI455X (gfx1250) — compile-verified
//
#include <hip/hip_runtime.h>

#define HDIM 512
#define WDIM 512
#define MAXITERS 100
#define EPSV 1e-6f

typedef __attribute__((ext_vector_type(2))) float v2f;
typedef __attribute__((ext_vector_type(8))) float v8f;

// ---------------------------------------------------------------------------
// Kernel 1: K = exp(-GAMMA*M), GAMMA=1.  HBM-bandwidth bound, float4 streams.
// ---------------------------------------------------------------------------
__global__ __launch_bounds__(256) void expneg_k(const float* __restrict__ M,
                                                float* __restrict__ K, int n4) {
  int i = blockIdx.x * blockDim.x + threadIdx.x;
  int stride = gridDim.x * blockDim.x;
  const float4* m4 = (const float4*)M;
  float4* k4 = (float4*)K;
  for (; i < n4; i += stride) {
    float4 m = m4[i];
    float4 r;
    r.x = expf(-m.x); r.y = expf(-m.y); r.z = expf(-m.z); r.w = expf(-m.w);
    k4[i] = r;
  }
}

// ---------------------------------------------------------------------------
// Kernel 2: persistent Sinkhorn. One 1024-thread block (32 waves) per batch.
// u,v,ktu in LDS; K stays L2-resident in d_out; grid sync per iteration via
// per-iteration arrive/converged counters in d_ws.
// ---------------------------------------------------------------------------
__global__ __launch_bounds__(1024) void sinkhorn_k(float* __restrict__ K,
                                                   unsigned* __restrict__ g_arr,
                                                   unsigned* __restrict__ g_conv,
                                                   int nblk) {
  __shared__ __align__(16) float u_lds[HDIM];
  __shared__ __align__(16) float v_lds[WDIM];
  __shared__ __align__(16) float t_lds[WDIM];   // ktu scratch
  __shared__ int s_conv;
  __shared__ int s_done;

  const int tid  = threadIdx.x;
  const int lane = tid & 31;
  const int wv   = tid >> 5;               // wave id 0..31
  float* Kb = K + (size_t)blockIdx.x * HDIM * WDIM;
  const float rmarg = 1.0f / HDIM;
  const float cmarg = 1.0f / WDIM;

  if (tid < WDIM) v_lds[tid] = 1.0f;
  if (tid == 0) s_done = 0;
  __syncthreads();

  for (int iter = 0; iter < MAXITERS; ++iter) {
    // ---- Phase A: u_h = rmarg / sum_w K[h][w]*v[w]  (wave-per-row, b128) ----
    if (tid == 0) s_conv = 1;
    {
      const float4* v4 = (const float4*)v_lds;
      for (int rr = 0; rr < 16; ++rr) {
        const int h = wv * 16 + rr;
        const float4* row4 = (const float4*)(Kb + (size_t)h * WDIM);
        float p = 0.f;
#pragma unroll
        for (int j = 0; j < 4; ++j) {
          float4 kk = row4[lane + 32 * j];
          float4 vv = v4[lane + 32 * j];
          p += kk.x * vv.x + kk.y * vv.y + kk.z * vv.z + kk.w * vv.w;
        }
#pragma unroll
        for (int off = 16; off >= 1; off >>= 1) p += __shfl_xor(p, off, 32);
        if (lane == 0) u_lds[h] = rmarg / p;
      }
    }
    __syncthreads();

    // ---- Phase B: ktu_w = sum_h u_h*K[h][w] via V_WMMA_F32_16X16X4_F32 ----
    // A (16x4) = u[h0..h0+3] replicated over M; per ISA layout lane L supplies
    //   VGPR0=u[h0+2*(L>>4)], VGPR1=u[h0+2*(L>>4)+1].
    // B (4x16) = K tile: lane L gives B[2*(L>>4)][L&15], B[2*(L>>4)+1][L&15]
    //   -> two coalesced row-major b32 loads per WMMA.
    // D[m][n] identical over m => every lane's c[0] = ktu_{w0+(lane&15)}.
    {
      const int w0 = wv * 16;
      const int hi = lane >> 4;
      const int n  = lane & 15;
      const float* col = Kb + (size_t)(2 * hi) * WDIM + (w0 + n);
      v8f c0 = {0.f, 0.f, 0.f, 0.f, 0.f, 0.f, 0.f, 0.f};
      v8f c1 = c0;
      for (int h0 = 0; h0 < HDIM; h0 += 8) {
        v2f a0, a1, b0, b1;
        a0.x = u_lds[h0 + 2 * hi];
        a0.y = u_lds[h0 + 2 * hi + 1];
        a1.x = u_lds[h0 + 4 + 2 * hi];
        a1.y = u_lds[h0 + 4 + 2 * hi + 1];
        const float* p0 = col + (size_t)h0 * WDIM;
        b0.x = p0[0];
        b0.y = p0[WDIM];
        b1.x = p0[4 * WDIM];
        b1.y = p0[5 * WDIM];
        c0 = __builtin_amdgcn_wmma_f32_16x16x4_f32(false, a0, false, b0,
                                                   (short)0, c0, false, false);
        c1 = __builtin_amdgcn_wmma_f32_16x16x4_f32(false, a1, false, b1,
                                                   (short)0, c1, false, false);
      }
      float acc = c0[0] + c1[0];
      if (lane < 16) {
        t_lds[w0 + lane] = acc;
        float beta = v_lds[w0 + lane] * acc;   // col marginal after row-norm
        if (fabsf(beta - cmarg) > EPSV) s_conv = 0;
      }
    }
    __syncthreads();

    // ---- Global convergence: arrive/converged counters, per-iteration ----
    if (tid == 0) {
      unsigned mc = (unsigned)s_conv;
      __hip_atomic_fetch_add(g_conv + iter, mc, __ATOMIC_RELAXED,
                             __HIP_MEMORY_SCOPE_AGENT);
      __hip_atomic_fetch_add(g_arr + iter, 1u, __ATOMIC_RELEASE,
                             __HIP_MEMORY_SCOPE_AGENT);
      while (__hip_atomic_load(g_arr + iter, __ATOMIC_ACQUIRE,
                               __HIP_MEMORY_SCOPE_AGENT) < (unsigned)nblk)
        __builtin_amdgcn_s_sleep(1);
      unsigned cv = __hip_atomic_load(g_conv + iter, __ATOMIC_RELAXED,
                                      __HIP_MEMORY_SCOPE_AGENT);
      s_done = (cv == (unsigned)nblk) ? 1 : 0;
    }
    __syncthreads();
    if (s_done) break;                 // break BEFORE col-normalize (ref semantics)
    if (tid < WDIM) v_lds[tid] = cmarg / t_lds[tid];
    __syncthreads();
  }

  __syncthreads();
  // ---- Finalize in place: P = u_h * K * v_w ----
  float4* P4 = (float4*)Kb;
  for (int i = tid; i < HDIM * WDIM / 4; i += 1024) {
    int h = i >> 7;
    int w = (i & 127) * 4;
    float uh = u_lds[h];
    float4 k = P4[i];
    k.x *= uh * v_lds[w];
    k.y *= uh * v_lds[w + 1];
    k.z *= uh * v_lds[w + 2];
    k.w *= uh * v_lds[w + 3];
    P4[i] = k;
  }
}

extern "C" void kernel_launch(void* const* d_in, const int* in_sizes, int n_in,
                              void* d_out, int out_size, void* d_ws, size_t ws_size,
                              hipStream_t stream) {
  const float* M = (const float*)d_in[0];
  float* K = (float*)d_out;                 // d_out doubles as the K buffer
  unsigned* arr  = (unsigned*)d_ws;         // arrive counters [128]
  unsigned* conv = arr + 128;               // converged counters [128]

  const int nelem = in_sizes[0];            // B*512*512
  const int nblk  = nelem / (HDIM * WDIM);  // B = 64

  hipMemsetAsync(d_ws, 0, 256 * sizeof(unsigned), stream);

  const int n4 = nelem / 4;
  int eb = (n4 + 255) / 256;
  if (eb > 4096) eb = 4096;
  expneg_k<<<eb, 256, 0, stream>>>(M, K, n4);
  sinkhorn_k<<<nblk, 1024, 0, stream>>>(K, arr, conv, nblk);
}